// Classifier_11690900980095
// MI455X (gfx1250) — compile-verified
//
#include <hip/hip_runtime.h>

#define NN 50000      // N_NODES
#define NE 800000     // N_EDGES
#define NG 128        // N_GRAPHS
#define HID 128       // HIDDEN
#define NC 4          // N_CLASSES

typedef float v2f __attribute__((ext_vector_type(2)));
typedef float v8f __attribute__((ext_vector_type(8)));

// ---------------- zero accumulators ----------------
__global__ void zero_kernel(float* __restrict__ p, int n) {
    int i = blockIdx.x * blockDim.x + threadIdx.x;
    int stride = gridDim.x * blockDim.x;
    for (; i < n; i += stride) p[i] = 0.0f;
}

// ---------------- degree counts ----------------
__global__ void degree_kernel(const int* __restrict__ src, const int* __restrict__ dst,
                              float* __restrict__ out_deg, float* __restrict__ in_deg) {
    int e = blockIdx.x * blockDim.x + threadIdx.x;
    if (e < NE) {
        atomicAdd(&out_deg[src[e]], 1.0f);
        atomicAdd(&in_deg[dst[e]], 1.0f);
    }
}

// ---------------- norms + layer-1 input (h = out_deg, scaled by norm_out) ----------------
__global__ void norm_kernel(const float* __restrict__ out_deg, const float* __restrict__ in_deg,
                            float* __restrict__ norm_out, float* __restrict__ norm_in,
                            float* __restrict__ s1) {
    int n = blockIdx.x * blockDim.x + threadIdx.x;
    if (n < NN) {
        float od = out_deg[n];
        float no = 1.0f / sqrtf(fmaxf(od, 1.0f));
        float ni = 1.0f / sqrtf(fmaxf(in_deg[n], 1.0f));
        norm_out[n] = no;
        norm_in[n]  = ni;
        s1[n]       = od * no;
    }
}

// ---------------- layer-1 scalar scatter: m1[dst] += s1[src] ----------------
__global__ void scatter1_kernel(const int* __restrict__ src, const int* __restrict__ dst,
                                const float* __restrict__ s1, float* __restrict__ m1) {
    int e = blockIdx.x * blockDim.x + threadIdx.x;
    if (e < NE) atomicAdd(&m1[dst[e]], s1[src[e]]);
}

// ---------------- layer-1 rank-1 "GEMM" + bias + relu, pre-scaled by norm_out for conv2 ----------------
__global__ void layer1_kernel(const float* __restrict__ m1, const float* __restrict__ norm_in,
                              const float* __restrict__ norm_out,
                              const float* __restrict__ W1, const float* __restrict__ b1,
                              float* __restrict__ h1s) {
    int idx = blockIdx.x * blockDim.x + threadIdx.x;  // NN*HID
    if (idx < NN * HID) {
        int n = idx >> 7;
        int j = idx & (HID - 1);
        float v = fmaf(m1[n] * norm_in[n], W1[j], b1[j]);
        v = fmaxf(v, 0.0f);
        h1s[idx] = v * norm_out[n];
    }
}

// ---------------- layer-2 feature scatter: one wave32 per edge, float4 per lane ----------------
__global__ void scatter2_kernel(const int* __restrict__ src, const int* __restrict__ dst,
                                const float* __restrict__ h1s, float* __restrict__ m2) {
    int wave = blockIdx.x * (blockDim.x >> 5) + (threadIdx.x >> 5);
    int lane = threadIdx.x & 31;
    if (wave < NE) {
        int s = src[wave];
        int d = dst[wave];
        const float4 v = *(const float4*)(h1s + (size_t)s * HID + lane * 4);
        float* o = m2 + (size_t)d * HID + lane * 4;
        atomicAdd(o + 0, v.x);
        atomicAdd(o + 1, v.y);
        atomicAdd(o + 2, v.z);
        atomicAdd(o + 3, v.w);
    }
}

// ---------------- layer-2 GEMM via V_WMMA_F32_16X16X4_F32 ----------------
// C[NN,128] = relu((m2 * norm_in) @ W2 + b2)
// One block = 8 waves; block b handles M-tile b, wave w handles N-tile w.
__global__ __launch_bounds__(256) void gemm2_wmma_kernel(
        const float* __restrict__ A,        // m2: NN x 128
        const float* __restrict__ norm_in,  // NN
        const float* __restrict__ W2,       // 128 x 128 row-major (K x N)
        const float* __restrict__ b2,       // 128
        float* __restrict__ h2) {           // NN x 128
    const int tileM = blockIdx.x;            // 0..NN/16-1
    const int wv    = threadIdx.x >> 5;      // 0..7 == N tile
    const int lane  = threadIdx.x & 31;
    const int half  = lane >> 4;             // 0: K={0,1}, 1: K={2,3} within a K-block
    const int l     = lane & 15;
    const int row   = tileM * 16 + l;        // A-matrix row this lane feeds
    const int col   = wv * 16 + l;           // B/C/D column this lane holds
    const float ani = norm_in[row];
    const float* __restrict__ arow = A + (size_t)row * HID;

    v8f c = {};
#pragma unroll 8
    for (int kb = 0; kb < HID / 4; ++kb) {
        const int k0 = kb * 4 + half * 2;
        v2f a, b;
        a.x = arow[k0]     * ani;            // A 16x4 layout: v0 = K0/K2, v1 = K1/K3
        a.y = arow[k0 + 1] * ani;
        b.x = W2[(size_t)(k0)     * HID + col];  // B 4x16: rows K striped the same way
        b.y = W2[(size_t)(k0 + 1) * HID + col];
        c = __builtin_amdgcn_wmma_f32_16x16x4_f32(
                /*neg_a=*/false, a, /*neg_b=*/false, b,
                /*c_mod=*/(short)0, c, /*reuse_a=*/false, /*reuse_b=*/false);
    }

    const float bias = b2[col];
#pragma unroll
    for (int v = 0; v < 8; ++v) {
        int r = tileM * 16 + half * 8 + v;   // C layout: VGPR v -> M = v (+8 for upper half-wave)
        float val = fmaxf(c[v] + bias, 0.0f);
        h2[(size_t)r * HID + col] = val;
    }
}

// ---------------- mean-pool accumulate ----------------
__global__ void pool_kernel(const float* __restrict__ h2, const int* __restrict__ gid,
                            float* __restrict__ gsum, float* __restrict__ gcnt) {
    int n = blockIdx.x;       // node
    int j = threadIdx.x;      // feature
    int g = gid[n];
    atomicAdd(&gsum[g * HID + j], h2[(size_t)n * HID + j]);
    if (j == 0) atomicAdd(&gcnt[g], 1.0f);
}

// ---------------- classifier: out = (gsum/cnt) @ Wc + bc ----------------
__global__ void classify_kernel(const float* __restrict__ gsum, const float* __restrict__ gcnt,
                                const float* __restrict__ Wc, const float* __restrict__ bc,
                                float* __restrict__ out) {
    int t = blockIdx.x * blockDim.x + threadIdx.x;
    if (t < NG * NC) {
        int g = t / NC;
        int cidx = t % NC;
        float acc = 0.0f;
        for (int j = 0; j < HID; ++j)
            acc = fmaf(gsum[g * HID + j], Wc[j * NC + cidx], acc);
        float cnt = fmaxf(gcnt[g], 1.0f);
        out[t] = acc / cnt + bc[cidx];
    }
}

extern "C" void kernel_launch(void* const* d_in, const int* in_sizes, int n_in,
                              void* d_out, int out_size, void* d_ws, size_t ws_size,
                              hipStream_t stream) {
    const int*   src = (const int*)d_in[0];
    const int*   dst = (const int*)d_in[1];
    const int*   gid = (const int*)d_in[2];
    const float* W1  = (const float*)d_in[3];
    const float* b1  = (const float*)d_in[4];
    const float* W2  = (const float*)d_in[5];
    const float* b2  = (const float*)d_in[6];
    const float* Wc  = (const float*)d_in[7];
    const float* bc  = (const float*)d_in[8];
    float* out = (float*)d_out;

    // Workspace layout: zeroed accumulators first (one contiguous region), then scratch.
    float* ws       = (float*)d_ws;
    float* out_deg  = ws;                          // NN
    float* in_deg   = out_deg + NN;                // NN
    float* m1       = in_deg + NN;                 // NN
    float* gsum     = m1 + NN;                     // NG*HID
    float* gcnt     = gsum + NG * HID;             // NG
    float* m2       = gcnt + NG;                   // NN*HID
    float* norm_out = m2 + (size_t)NN * HID;       // NN
    float* norm_in  = norm_out + NN;               // NN
    float* s1       = norm_in + NN;                // NN
    float* h1s      = s1 + NN;                     // NN*HID (reused as h2 after scatter2)

    const int zeroN = 3 * NN + NG * HID + NG + NN * HID;
    zero_kernel<<<2048, 256, 0, stream>>>(ws, zeroN);

    degree_kernel<<<(NE + 255) / 256, 256, 0, stream>>>(src, dst, out_deg, in_deg);
    norm_kernel<<<(NN + 255) / 256, 256, 0, stream>>>(out_deg, in_deg, norm_out, norm_in, s1);
    scatter1_kernel<<<(NE + 255) / 256, 256, 0, stream>>>(src, dst, s1, m1);
    layer1_kernel<<<(NN * HID + 255) / 256, 256, 0, stream>>>(m1, norm_in, norm_out, W1, b1, h1s);
    scatter2_kernel<<<(NE + 7) / 8, 256, 0, stream>>>(src, dst, h1s, m2);
    gemm2_wmma_kernel<<<NN / 16, 256, 0, stream>>>(m2, norm_in, W2, b2, h1s /* = h2 */);
    pool_kernel<<<NN, HID, 0, stream>>>(h1s, gid, gsum, gcnt);
    classify_kernel<<<(NG * NC + 255) / 256, 256, 0, stream>>>(gsum, gcnt, Wc, bc, out);
}